// Conv_83133386981494
// MI455X (gfx1250) — compile-verified
//
#include <hip/hip_runtime.h>

// ---------------- problem constants (from the reference) ----------------
#define V_    2562
#define DEG_  8
#define FIN_  8
#define FOUT_ 16
#define K_    5
#define S_    512                 // B*X*Y*Z = 1*8*8*8
#define D_    (FIN_ * S_)         // 4096-float feature row per vertex
#define KDIM_ (K_ * FIN_)         // 40 = contraction depth of final GEMM

typedef float v2f __attribute__((ext_vector_type(2)));
typedef float v4f __attribute__((ext_vector_type(4)));
typedef float v8f __attribute__((ext_vector_type(8)));

// ---------------------------------------------------------------
// Kernel 1: build cheb[0] = x0[v, f*512+s] = inputs[f, v, s]
// one block per (f,v) chunk; 128 threads x float4 = 512 floats
// ---------------------------------------------------------------
__global__ __launch_bounds__(128) void cheb0_transpose_kernel(
    const float* __restrict__ in, float* __restrict__ cheb0) {
  int chunk = blockIdx.x;            // 0 .. FIN_*V_-1
  int f = chunk / V_;
  int v = chunk % V_;
  const v4f* __restrict__ src = (const v4f*)(in + ((size_t)f * V_ + v) * S_);
  v4f* __restrict__ dst = (v4f*)(cheb0 + (size_t)v * D_ + (size_t)f * S_);
  dst[threadIdx.x] = src[threadIdx.x];
}

// ---------------------------------------------------------------
// Kernel 2: SpMV + Chebyshev recurrence, one block per vertex row.
// cheb_mode==0: xout = L @ xin            (k == 1)
// cheb_mode==1: xout = 2*(L @ xin) - xm2  (k >= 2)
// ---------------------------------------------------------------
__global__ __launch_bounds__(256) void spmv_cheb_kernel(
    const int*   __restrict__ cols,
    const float* __restrict__ vals,
    const float* __restrict__ xin,
    const float* __restrict__ xm2,
    float*       __restrict__ xout,
    int cheb_mode) {
  const int v = blockIdx.x;
  const int base = v * DEG_;

  int   c[DEG_];
  float w[DEG_];
#pragma unroll
  for (int e = 0; e < DEG_; ++e) {
    c[e] = cols[base + e];           // uniform per block -> scalarized
    w[e] = vals[base + e];
  }
#pragma unroll
  for (int e = 0; e < DEG_; ++e)     // warm the gathered rows (global_prefetch_b8)
    __builtin_prefetch(xin + (size_t)c[e] * D_ + 4 * threadIdx.x, 0, 0);

  const v4f* __restrict__ xr[DEG_];
#pragma unroll
  for (int e = 0; e < DEG_; ++e)
    xr[e] = (const v4f*)(xin + (size_t)c[e] * D_);

  const v4f* __restrict__ pm2 = cheb_mode ? (const v4f*)(xm2 + (size_t)v * D_) : nullptr;
  v4f* __restrict__ po = (v4f*)(xout + (size_t)v * D_);

  for (int d4 = threadIdx.x; d4 < D_ / 4; d4 += blockDim.x) {
    v4f acc = (v4f)(0.0f);
#pragma unroll
    for (int e = 0; e < DEG_; ++e)
      acc += w[e] * xr[e][d4];
    if (cheb_mode)
      acc = 2.0f * acc - pm2[d4];
    po[d4] = acc;
  }
}

// ---------------------------------------------------------------
// Kernel 3: out[o, v, s] = sum_{j<40} cheb[j/8][v][(j%8)*512+s] * W[j, o] + bias[o]
// via V_WMMA_F32_16X16X4_F32. One wave (32 lanes) per 16-row tile.
// M tile = 16 consecutive s within one vertex (512 % 16 == 0), N = FOUT = 16,
// Kdim = 40 -> 10 chained k=4 WMMAs, accumulator carried in C.
// ---------------------------------------------------------------
__global__ __launch_bounds__(256) void cheb_gemm_wmma_kernel(
    const float* __restrict__ cheb,    // [K_][V_][D_]
    const float* __restrict__ weight,  // [KDIM_][FOUT_] flattened [K,FIN,FOUT]
    const float* __restrict__ bias,    // [FOUT_]
    float*       __restrict__ out) {   // [FOUT_][V_][S_]
  const int wave = threadIdx.x >> 5;
  const int lane = threadIdx.x & 31;
  const long ntiles = (long)V_ * S_ / 16;  // 81984
  const long tile = (long)blockIdx.x * 8 + wave;
  if (tile >= ntiles) return;

  const int mb = (int)(tile * 16);
  const int v  = mb / S_;
  const int sb = mb % S_;
  const int hi  = lane >> 4;     // half-wave select (K offset +2 per ISA layout)
  const int l16 = lane & 15;     // A: row M;  B/C/D: column N

  // --- B fragments for all 10 k-chunks (weight is 640 floats, stays in L0/L1)
  // layout: VGPR0 = rows K=0 (lanes 0-15) / K=2 (lanes 16-31), VGPR1 = K=1 / K=3
  v2f bfrag[10];
#pragma unroll
  for (int cblk = 0; cblk < 10; ++cblk) {
    const int kc = cblk * 4 + hi * 2;
    bfrag[cblk].x = weight[(kc + 0) * FOUT_ + l16];
    bfrag[cblk].y = weight[(kc + 1) * FOUT_ + l16];
  }

  // --- C init with bias (every row of column N gets bias[N])
  v8f acc;
  const float bv = bias[l16];
#pragma unroll
  for (int r = 0; r < 8; ++r) acc[r] = bv;

  // --- 10 chained WMMAs over the 40-deep contraction
#pragma unroll
  for (int cblk = 0; cblk < 10; ++cblk) {
    const int j0    = cblk * 4;      // chunks never straddle a k boundary (8 % 4 == 0)
    const int kcheb = j0 >> 3;
    const int f0    = j0 & 7;
    const float* __restrict__ abase =
        cheb + ((size_t)kcheb * V_ + v) * D_ + (size_t)sb + l16;
    const int kk = hi * 2;           // lanes 0-15: K 0/1, lanes 16-31: K 2/3
    v2f a;
    a.x = abase[(size_t)(f0 + kk + 0) * S_];
    a.y = abase[(size_t)(f0 + kk + 1) * S_];
    acc = __builtin_amdgcn_wmma_f32_16x16x4_f32(
        /*neg_a=*/false, a, /*neg_b=*/false, bfrag[cblk],
        /*c_mod=*/(short)0, acc, /*reuse_a=*/false, /*reuse_b=*/false);
  }

  // --- store D per 16x16 f32 layout: VGPR r -> M = r + 8*hi, N = l16
#pragma unroll
  for (int r = 0; r < 8; ++r) {
    const int m = (hi << 3) + r;
    out[((size_t)l16 * V_ + v) * S_ + (size_t)(sb + m)] = acc[r];
  }
}

// ---------------------------------------------------------------
// launcher
// d_in: 0 inputs f32 [1,FIN,V,8,8,8], 1 lap_rows i32 (unused: rows are v=i/8 by
// construction), 2 lap_cols i32, 3 lap_vals f32, 4 weight f32 [K,FIN,FOUT],
// 5 bias f32 [FOUT].  d_ws: cheb stack, K_*V_*D_ floats (~200 MB).
// ---------------------------------------------------------------
extern "C" void kernel_launch(void* const* d_in, const int* in_sizes, int n_in,
                              void* d_out, int out_size, void* d_ws, size_t ws_size,
                              hipStream_t stream) {
  (void)in_sizes; (void)n_in; (void)out_size; (void)ws_size;
  const float* inputs = (const float*)d_in[0];
  const int*   cols   = (const int*)  d_in[2];
  const float* vals   = (const float*)d_in[3];
  const float* weight = (const float*)d_in[4];
  const float* bias   = (const float*)d_in[5];
  float* out  = (float*)d_out;
  float* cheb = (float*)d_ws;
  const size_t rowblk = (size_t)V_ * D_;

  // x0
  cheb0_transpose_kernel<<<FIN_ * V_, 128, 0, stream>>>(inputs, cheb);
  // x1 = L @ x0
  spmv_cheb_kernel<<<V_, 256, 0, stream>>>(cols, vals, cheb, nullptr,
                                           cheb + rowblk, 0);
  // x_k = 2 L x_{k-1} - x_{k-2}, k = 2..4
  for (int k = 2; k < K_; ++k)
    spmv_cheb_kernel<<<V_, 256, 0, stream>>>(cols, vals,
                                             cheb + (size_t)(k - 1) * rowblk,
                                             cheb + (size_t)(k - 2) * rowblk,
                                             cheb + (size_t)k * rowblk, 1);
  // out = cheb (M=V*512, Kdim=40) @ W (40,16) + bias, WMMA f32
  const int ntiles = V_ * S_ / 16;            // 81984
  const int nblocks = ntiles / 8;             // 8 waves / block of 256
  cheb_gemm_wmma_kernel<<<nblocks, 256, 0, stream>>>(cheb, weight, bias, out);
}